// SpatialPositionalEncoding_7043746365726
// MI455X (gfx1250) — compile-verified
//
#include <hip/hip_runtime.h>
#include <hip/hip_bf16.h>

typedef __attribute__((ext_vector_type(2))) float v2f;
typedef __attribute__((ext_vector_type(4))) float v4f;
typedef __attribute__((ext_vector_type(8))) float v8f;

#define Bdim 4
#define Ndim 1024
#define Fdim 16
#define Hdim 32
#define NEGV -1000000000.0f
#define THRESH 1e-6f

// One workgroup (8 wave32) per (b,n) row: WMMA-based MLP scores for 1024 m's,
// masked softmax in LDS, dense weights to d_ws.
__global__ __launch_bounds__(256) void score_softmax_kernel(
    const float* __restrict__ path_attr,   // (B,N,N,F)
    const int*   __restrict__ mask_down,   // (B,N,N)
    const int*   __restrict__ mask_up,     // (B,N,N)
    const float* __restrict__ W1,          // (F,H)
    const float* __restrict__ b1,          // (H)
    const float* __restrict__ W2,          // (H,1)
    const float* __restrict__ b2,          // (1)
    float*       __restrict__ wdense)      // (B,N,N)
{
    __shared__ float s_scores[Ndim];
    __shared__ float s_red[256];

    const int bn   = blockIdx.x;       // 0 .. B*N-1
    const int tid  = threadIdx.x;
    const int lane = tid & 31;
    const int wave = tid >> 5;
    const int lo   = lane & 15;        // column / M-within-half index
    const int hi   = lane >> 4;        // K-half selector (memory k = 8*hi + p)

    // K-permuted mapping: WMMA step j, A-operand VGPR v, lane-half h holds
    // memory k = 2*j + v + 8*h.  B operand must use the same permutation:
    v2f bw[2][4];
    #pragma unroll
    for (int c = 0; c < 2; ++c) {
        const int c0 = c * 16;
        #pragma unroll
        for (int j = 0; j < 4; ++j) {
            bw[c][j].x = W1[(2 * j + 0 + 8 * hi) * Hdim + c0 + lo];
            bw[c][j].y = W1[(2 * j + 1 + 8 * hi) * Hdim + c0 + lo];
        }
    }
    const float b1A = b1[lo], b1B = b1[lo + 16];
    const float w2A = W2[lo], w2B = W2[lo + 16];
    const float b2v = b2[0];

    const float* rowbase = path_attr + (size_t)bn * Ndim * Fdim;

    // 64 tiles of 16 m's; 8 waves -> 8 tiles per wave, software-pipelined.
    // Each lane streams 2 contiguous 16B chunks of its row (non-temporal).
    const float* xptr = rowbase + (size_t)(wave * 16 + lo) * Fdim + 8 * hi;
    v4f a0 = __builtin_nontemporal_load((const v4f*)(xptr));
    v4f a1 = __builtin_nontemporal_load((const v4f*)(xptr + 4));

    for (int t = wave; t < Ndim / 16; t += 8) {
        const int m0 = t * 16;

        // Prefetch next tile's A while this tile computes/reduces.
        const int tn = (t + 8 < Ndim / 16) ? (t + 8) : t;
        const float* xnext = rowbase + (size_t)(tn * 16 + lo) * Fdim + 8 * hi;
        v4f na0 = __builtin_nontemporal_load((const v4f*)(xnext));
        v4f na1 = __builtin_nontemporal_load((const v4f*)(xnext + 4));

        v8f acc0 = {}, acc1 = {};
        #pragma unroll
        for (int j = 0; j < 4; ++j) {
            v2f a;
            a.x = (j < 2) ? a0[2 * j]     : a1[2 * j - 4];
            a.y = (j < 2) ? a0[2 * j + 1] : a1[2 * j - 3];
            acc0 = __builtin_amdgcn_wmma_f32_16x16x4_f32(
                false, a, false, bw[0][j], (short)0, acc0, false, false);
            acc1 = __builtin_amdgcn_wmma_f32_16x16x4_f32(
                false, a, false, bw[1][j], (short)0, acc1, false, false);
        }

        // acc{0,1}[r] = h[m0 + r + 8*hi][(c0)+lo].  Bias, lrelu, dot with W2
        // across the 16 h-columns held by each 16-lane half.
        float s[8];
        #pragma unroll
        for (int r = 0; r < 8; ++r) {
            float hA = acc0[r] + b1A;
            float hB = acc1[r] + b1B;
            hA = fmaxf(hA, 0.1f * hA);     // leaky_relu slope 0.1
            hB = fmaxf(hB, 0.1f * hB);
            s[r] = hA * w2A + hB * w2B;
        }
        #pragma unroll
        for (int r = 0; r < 8; ++r) {
            #pragma unroll
            for (int msk = 1; msk < 16; msk <<= 1)
                s[r] += __shfl_xor(s[r], msk, 32);
        }
        if (lo == 0) {
            #pragma unroll
            for (int r = 0; r < 8; ++r)
                s_scores[m0 + 8 * hi + r] = s[r] + b2v;
        }
        a0 = na0;
        a1 = na1;
    }
    __syncthreads();

    // Masked softmax over the 1024 scores (4 elements per thread).
    const int* mdrow = mask_down + (size_t)bn * Ndim;
    const int* murow = mask_up   + (size_t)bn * Ndim;

    bool  msk[4];
    float sc[4];
    float lmax = -3.4e38f;
    #pragma unroll
    for (int k = 0; k < 4; ++k) {
        const int m = tid + k * 256;
        const int md = __builtin_nontemporal_load(mdrow + m);
        const int mu = __builtin_nontemporal_load(murow + m);
        const bool mk = (md > 0) && (mu > 0);
        const float v = mk ? s_scores[m] : NEGV;
        msk[k] = mk;
        sc[k]  = v;
        lmax = fmaxf(lmax, v);
    }
    s_red[tid] = lmax;
    __syncthreads();
    #pragma unroll
    for (int off = 128; off > 0; off >>= 1) {
        if (tid < off) s_red[tid] = fmaxf(s_red[tid], s_red[tid + off]);
        __syncthreads();
    }
    const float gmax = s_red[0];
    __syncthreads();

    float ex[4];
    float lsum = 0.0f;
    #pragma unroll
    for (int k = 0; k < 4; ++k) {
        ex[k] = __expf(sc[k] - gmax);
        lsum += ex[k];
    }
    s_red[tid] = lsum;
    __syncthreads();
    #pragma unroll
    for (int off = 128; off > 0; off >>= 1) {
        if (tid < off) s_red[tid] += s_red[tid + off];
        __syncthreads();
    }
    const float rinv = 1.0f / s_red[0];

    float* wrow = wdense + (size_t)bn * Ndim;
    #pragma unroll
    for (int k = 0; k < 4; ++k) {
        const int m = tid + k * 256;
        float w = ex[k] * rinv;
        w = (msk[k] && (w >= THRESH)) ? w : 0.0f;
        wrow[m] = w;
    }
}

// out[0 .. 2E)            = edge_index (src then tgt), as float values
// out[2E .. 2E + B*E)     = edge_weight[b, e] = wdense[b, src[e], tgt[e]]
__global__ void gather_kernel(const int* __restrict__ src,
                              const int* __restrict__ tgt,
                              const float* __restrict__ wdense,
                              float* __restrict__ out, int E)
{
    const int t = blockIdx.x * blockDim.x + threadIdx.x;
    const int total = 2 * E + Bdim * E;
    if (t >= total) return;
    if (t < 2 * E) {
        out[t] = (t < E) ? (float)src[t] : (float)tgt[t - E];
    } else {
        const int u = t - 2 * E;
        const int b = u / E;
        const int e = u - b * E;
        const size_t idx = ((size_t)b * Ndim + (size_t)src[e]) * Ndim + (size_t)tgt[e];
        out[t] = wdense[idx];
    }
}

extern "C" void kernel_launch(void* const* d_in, const int* in_sizes, int n_in,
                              void* d_out, int out_size, void* d_ws, size_t ws_size,
                              hipStream_t stream) {
    const int*   mask_down = (const int*)d_in[0];
    const int*   mask_up   = (const int*)d_in[1];
    const float* path_attr = (const float*)d_in[2];
    const int*   src       = (const int*)d_in[3];
    const int*   tgt       = (const int*)d_in[4];
    const float* W1        = (const float*)d_in[5];
    const float* b1        = (const float*)d_in[6];
    const float* W2        = (const float*)d_in[7];
    const float* b2        = (const float*)d_in[8];
    float* out    = (float*)d_out;
    float* wdense = (float*)d_ws;   // B*N*N floats = 16 MB dense weights
    const int E = in_sizes[3];

    score_softmax_kernel<<<Bdim * Ndim, 256, 0, stream>>>(
        path_attr, mask_down, mask_up, W1, b1, W2, b2, wdense);

    const int total = 2 * E + Bdim * E;
    if (total > 0) {
        gather_kernel<<<(total + 255) / 256, 256, 0, stream>>>(
            src, tgt, wdense, out, E);
    }
}